// Q2TL_20005957665175
// MI455X (gfx1250) — compile-verified
//
#include <hip/hip_runtime.h>
#include <hip/hip_bf16.h>
#include <stdint.h>

#define THREADS_A 256
#define THREADS_B 256
#define CONF_THRES 0.2f

typedef __attribute__((ext_vector_type(2))) float v2f;
typedef __attribute__((ext_vector_type(8))) float v8f;

// Low 32 bits of a flat pointer into LDS == LDS byte offset (gfx1250 aperture:
// LDS offset is the low dword), which is what async-to-LDS VDST expects.
__device__ __forceinline__ uint32_t lds_off_of(const void* p) {
    return (uint32_t)(uintptr_t)p;
}

// CDNA5 async global->LDS copies (ASYNCcnt-tracked), per ISA 15.18.3 op 96/98.
__device__ __forceinline__ void async_g2l_b128(uint32_t lds_addr, const void* gaddr) {
    asm volatile("global_load_async_to_lds_b128 %0, %1, off"
                 :: "v"(lds_addr), "v"((uint64_t)(uintptr_t)gaddr)
                 : "memory");
}
__device__ __forceinline__ void async_g2l_b32(uint32_t lds_addr, const void* gaddr) {
    asm volatile("global_load_async_to_lds_b32 %0, %1, off"
                 :: "v"(lds_addr), "v"((uint64_t)(uintptr_t)gaddr)
                 : "memory");
}
__device__ __forceinline__ void wait_async0() {
    asm volatile("s_wait_asynccnt 0x0" ::: "memory");
}

// ---------------------------------------------------------------------------
// Kernel A: per row of x[B,S]: rowmax, log-sum-exp, argmax (pseudo), x[b,target]
// Row is staged HBM -> LDS once via async global->LDS loads; both passes read LDS.
// ---------------------------------------------------------------------------
__global__ void __launch_bounds__(THREADS_A)
rowstats_x_kernel(const float* __restrict__ x,
                  const int* __restrict__ target, int S,
                  float* __restrict__ row_max, float* __restrict__ row_lse,
                  int* __restrict__ row_pseudo, float* __restrict__ x_at_tgt) {
    extern __shared__ float4 smem4[];
    float* smem = (float*)smem4;
    __shared__ float red_f[16];
    __shared__ int   red_i[16];
    __shared__ float bc[4];

    const int row = blockIdx.x;
    const int tid = threadIdx.x;
    const unsigned lane = tid & 31u, wid = tid >> 5;
    const int nwave = THREADS_A / 32;
    const float* xrow = x + (size_t)row * (size_t)S;

    // --- async copy row -> LDS (16B per lane-op) ---
    const int nvec = S >> 2;
    const int rem  = S & 3;
    for (int c = tid; c < nvec; c += THREADS_A)
        async_g2l_b128(lds_off_of(&smem4[c]), xrow + 4 * c);
    if (tid < rem) {
        int i = nvec * 4 + tid;
        async_g2l_b32(lds_off_of(&smem[i]), xrow + i);
    }
    wait_async0();
    __syncthreads();

    // --- pass 1: max + argmax (first-occurrence tie-break) ---
    float m = -__builtin_inff(); int mi = 0x7fffffff;
    for (int c = tid; c < nvec; c += THREADS_A) {
        float4 v = smem4[c];
        int b = 4 * c;
        if (v.x > m) { m = v.x; mi = b + 0; }
        if (v.y > m) { m = v.y; mi = b + 1; }
        if (v.z > m) { m = v.z; mi = b + 2; }
        if (v.w > m) { m = v.w; mi = b + 3; }
    }
    if (tid < rem) {
        int i = nvec * 4 + tid;
        float v = smem[i];
        if (v > m) { m = v; mi = i; }
    }
    for (int off = 16; off > 0; off >>= 1) {
        float om = __shfl_down(m, off);
        int   oi = __shfl_down(mi, off);
        if (om > m || (om == m && oi < mi)) { m = om; mi = oi; }
    }
    if (lane == 0) { red_f[wid] = m; red_i[wid] = mi; }
    __syncthreads();
    if (tid == 0) {
        for (int w = 1; w < nwave; ++w)
            if (red_f[w] > m || (red_f[w] == m && red_i[w] < mi)) { m = red_f[w]; mi = red_i[w]; }
        bc[0] = m;
    }
    __syncthreads();
    const float rowm = bc[0];

    // --- pass 2: sum of exp(v - rowmax) ---
    float s = 0.f;
    for (int c = tid; c < nvec; c += THREADS_A) {
        float4 v = smem4[c];
        s += __expf(v.x - rowm) + __expf(v.y - rowm) +
             __expf(v.z - rowm) + __expf(v.w - rowm);
    }
    if (tid < rem) s += __expf(smem[nvec * 4 + tid] - rowm);
    for (int off = 16; off > 0; off >>= 1) s += __shfl_down(s, off);
    __syncthreads();                    // red_f reuse safe: prior reads done
    if (lane == 0) red_f[wid] = s;
    __syncthreads();
    if (tid == 0) {
        float tot = 0.f;
        for (int w = 0; w < nwave; ++w) tot += red_f[w];
        row_max[row]    = rowm;
        row_lse[row]    = __logf(tot);
        row_pseudo[row] = mi;
        x_at_tgt[row]   = smem[target[row]];   // gather from LDS, free
    }
}

// ---------------------------------------------------------------------------
// Kernel B: per row of x_main[B,M]: softmax stats in registers, gather the two
// main-class probs, blend lambdas, emit loss[b].
// ---------------------------------------------------------------------------
__global__ void __launch_bounds__(THREADS_B)
rowloss_kernel(const float* __restrict__ xm,
               const int* __restrict__ target, const int* __restrict__ sub2main,
               int M,
               const float* __restrict__ row_max, const float* __restrict__ row_lse,
               const int* __restrict__ row_pseudo, const float* __restrict__ x_at_tgt,
               float* __restrict__ loss) {
    __shared__ float red_f[16];
    __shared__ float bc[2];
    const int row = blockIdx.x;
    const int tid = threadIdx.x;
    const unsigned lane = tid & 31u, wid = tid >> 5;
    const int nwave = THREADS_B / 32;
    const float* xr = xm + (size_t)row * (size_t)M;

    float vals[8]; int n = 0;
    float m = -__builtin_inff();
    for (int i = tid; i < M && n < 8; i += THREADS_B) {
        float v = xr[i];
        vals[n++] = v;
        m = fmaxf(m, v);
    }
    for (int off = 16; off > 0; off >>= 1) m = fmaxf(m, __shfl_down(m, off));
    if (lane == 0) red_f[wid] = m;
    __syncthreads();
    if (tid == 0) {
        float mm = red_f[0];
        for (int w = 1; w < nwave; ++w) mm = fmaxf(mm, red_f[w]);
        bc[0] = mm;
    }
    __syncthreads();
    const float rm = bc[0];

    float s = 0.f;
    for (int k = 0; k < n; ++k) s += __expf(vals[k] - rm);
    for (int off = 16; off > 0; off >>= 1) s += __shfl_down(s, off);
    __syncthreads();
    if (lane == 0) red_f[wid] = s;
    __syncthreads();
    if (tid == 0) {
        float ssum = 0.f;
        for (int w = 0; w < nwave; ++w) ssum += red_f[w];
        const int t  = target[row];
        const int p  = row_pseudo[row];
        const int jt = sub2main[t];
        const int jp = sub2main[p];
        const float pt = __expf(xr[jt] - rm) / ssum;
        const float pp = __expf(xr[jp] - rm) / ssum;
        const bool corr = (t != p) && (pp >= CONF_THRES);
        float lam_t = 1.f, lam_c = 0.f;
        if (corr) { float inv = 1.f / (pt + pp); lam_t = pt * inv; lam_c = pp * inv; }
        const float logp_t = x_at_tgt[row] - row_max[row] - row_lse[row];
        const float logp_p = -row_lse[row];          // x[b,pseudo]==rowmax
        loss[row] = -(lam_t * logp_t + lam_c * logp_p);
    }
}

// ---------------------------------------------------------------------------
// Kernel C: mean over loss[B] with a single wave. 32 lane partials are
// collapsed 32->16 by one v_wmma_f32_16x16x4_f32 (ones A-matrix sums along K:
// A(m,k)=1, B(0,n)=p_n, B(2,n)=p_{n+16} => D(m,n)=p_n+p_{n+16}), then 4
// xor-shuffles finish. EXEC is all-1s (full wave, uniform flow) as WMMA needs.
// ---------------------------------------------------------------------------
__global__ void __launch_bounds__(32)
mean_kernel(const float* __restrict__ loss, int B, float* __restrict__ out) {
    const int lane = threadIdx.x;
    float p = 0.f;
    for (int i = lane; i < B; i += 32) p += loss[i];

    v2f a; a[0] = 1.0f; a[1] = 1.0f;   // A = ones(16x4)
    v2f b; b[0] = p;    b[1] = 0.0f;   // B(4x16): K=0/2 rows carry partials
    v8f c = {};
    v8f d = __builtin_amdgcn_wmma_f32_16x16x4_f32(
        /*neg_a=*/false, a, /*neg_b=*/false, b,
        /*c_mod=*/(short)0, c, /*reuse_a=*/false, /*reuse_b=*/false);

    float v = d[0];                    // = p_n + p_{n+16}, n = lane % 16
    v += __shfl_xor(v, 8);
    v += __shfl_xor(v, 4);
    v += __shfl_xor(v, 2);
    v += __shfl_xor(v, 1);
    if (lane == 0) out[0] = v / (float)B;
}

// ---------------------------------------------------------------------------
extern "C" void kernel_launch(void* const* d_in, const int* in_sizes, int n_in,
                              void* d_out, int out_size, void* d_ws, size_t ws_size,
                              hipStream_t stream) {
    const float* x        = (const float*)d_in[0];
    const float* x_main   = (const float*)d_in[1];
    const int*   target   = (const int*)d_in[2];
    const int*   sub2main = (const int*)d_in[3];

    const int B = in_sizes[2];          // 4096
    const int S = in_sizes[3];          // 10000
    const int M = in_sizes[1] / B;      // 1000

    // ws layout: 5 arrays of B floats/ints = 80 KB for B=4096.
    float* row_max = (float*)d_ws;
    float* row_lse = row_max + B;
    float* x_at    = row_lse + B;
    float* loss    = x_at + B;
    int*   pseudo  = (int*)(loss + B);

    const size_t shmem = ((size_t)S * 4 + 15) & ~(size_t)15;   // row staged in LDS

    rowstats_x_kernel<<<B, THREADS_A, shmem, stream>>>(
        x, target, S, row_max, row_lse, pseudo, x_at);
    rowloss_kernel<<<B, THREADS_B, 0, stream>>>(
        x_main, target, sub2main, M, row_max, row_lse, pseudo, x_at, loss);
    mean_kernel<<<1, 32, 0, stream>>>(loss, B, (float*)d_out);
}